// DotProductAttentionD_27891517620908
// MI455X (gfx1250) — compile-verified
//
#include <hip/hip_runtime.h>
#include <hip/hip_bf16.h>

// CDNA5 (gfx1250) fused attention with materialized attn matrix.
// out  = softmax(mask ? QK^T : -1e9) @ V   [16,4096,64] f32
// attn = softmax probabilities              [16,4096,4096] f32 (1.07 GB -> HBM-bound)
//
// fp32 WMMA (V_WMMA_F32_16X16X4_F32) everywhere: the kernel is bound by the
// 1.07 GB attn write (~46us @ 23.3 TB/s); fp32 matrix cores (~70 GFLOP total)
// roughly match that, so lower precision would not help and would diverge from
// the fp32 reference.

typedef __attribute__((ext_vector_type(2))) float v2f;
typedef __attribute__((ext_vector_type(8))) float v8f;

#define HDIM 16
#define SEQ  4096
#define DDIM 64
#define NEGV (-1.0e9f)

#define ROWS 16          // query rows per workgroup
#define NWAVE 8          // 256 threads = 8 wave32
#define SC_STRIDE 4100   // scores row stride (floats); 4100%64=4 -> conflict-free frag reads
#define ST_STRIDE 68     // K/V staging row stride (floats); 68%64=4
#define CT_PER_WAVE 32   // 256 column tiles / 8 waves

__global__ __launch_bounds__(256) void attn_fused_kernel(
    const float* __restrict__ q, const float* __restrict__ k,
    const float* __restrict__ v, const int* __restrict__ mask,
    float* __restrict__ out, float* __restrict__ attn)
{
    extern __shared__ float smem[];
    float* sc      = smem;                               // [16][4100] scores
    float* stg     = sc + ROWS * SC_STRIDE;              // [8][16][68] per-wave K/V staging
    float* red     = stg + NWAVE * 16 * ST_STRIDE;       // [16][16] partial reduce
    float* rowstat = red + 256;                          // [16] row max / 1/sum

    const int h    = blockIdx.x;        // head (x-major so heads share mask tiles in L2)
    const int rt   = blockIdx.y;        // 16-row tile
    const int tid  = threadIdx.x;
    const int wave = tid >> 5;
    const int lane = tid & 31;
    const int half = lane >> 4;         // K-pair selector within WMMA fragment
    const int ln16 = lane & 15;         // M (A) or N (B/C/D) index

    const size_t headOff = (size_t)h * SEQ * DDIM;
    const float* qh = q + headOff + (size_t)rt * ROWS * DDIM;
    const float* kh = k + headOff;
    const float* vh = v + headOff;
    float*    mystg = stg + wave * 16 * ST_STRIDE;

    // ---- Q A-fragments for all 16 K=4 chunks, resident for whole kernel ----
    // A layout (ISA 7.12.2, 32-bit 16x4): lane L: M = L&15, holds K = {2*half, 2*half+1}
    v2f qfrag[16];
    {
        const float* qrow = qh + (size_t)ln16 * DDIM;
        #pragma unroll
        for (int c = 0; c < 16; ++c) {
            int col = 4 * c + 2 * half;
            qfrag[c].x = qrow[col];
            qfrag[c].y = qrow[col + 1];
        }
    }

    // ================= Phase 1: S = Q K^T (masked) -> LDS =================
    for (int ct = wave * CT_PER_WAVE; ct < wave * CT_PER_WAVE + CT_PER_WAVE; ++ct) {
        // stage K tile [16 seq rows][64 dims], coalesced float4 loads
        {
            const float4* src = (const float4*)(kh + (size_t)ct * 16 * DDIM);
            #pragma unroll
            for (int i = 0; i < 8; ++i) {
                int j = lane + 32 * i;            // float4 index 0..255
                int r = j >> 4, c4 = j & 15;
                float4 d = src[j];
                float* dst = mystg + r * ST_STRIDE + c4 * 4;
                dst[0] = d.x; dst[1] = d.y; dst[2] = d.z; dst[3] = d.w;
            }
        }
        v8f acc = {};
        #pragma unroll
        for (int c = 0; c < 16; ++c) {
            // B = K^T chunk: B[kk][n] = K[ct*16+n][4c+kk]; lane: N=ln16, K pair by half
            v2f b;
            const float* bp = mystg + ln16 * ST_STRIDE + 4 * c + 2 * half;
            b.x = bp[0]; b.y = bp[1];
            acc = __builtin_amdgcn_wmma_f32_16x16x4_f32(
                false, qfrag[c], false, b, (short)0, acc, false, false);
        }
        // mask + spill scores to LDS. C/D layout: vgpr j -> M=j+8*half, N=ln16
        #pragma unroll
        for (int j = 0; j < 8; ++j) {
            int row = j + 8 * half;
            int col = ct * 16 + ln16;
            int mv  = mask[(size_t)(rt * ROWS + row) * SEQ + col];
            sc[row * SC_STRIDE + col] = mv ? acc[j] : NEGV;
        }
    }
    __syncthreads();

    // ================= Phase 2: softmax over 4096 columns =================
    const int srow  = tid & 15;
    const int scol0 = tid >> 4;
    float* myrow = sc + srow * SC_STRIDE;

    float m = -3.4e38f;
    for (int c = scol0; c < SEQ; c += 16) m = fmaxf(m, myrow[c]);
    red[srow * 16 + scol0] = m;
    __syncthreads();
    if (tid < 16) {
        float mm = red[tid * 16];
        #pragma unroll
        for (int i = 1; i < 16; ++i) mm = fmaxf(mm, red[tid * 16 + i]);
        rowstat[tid] = mm;
    }
    __syncthreads();
    float rmax = rowstat[srow];
    float s = 0.f;
    for (int c = scol0; c < SEQ; c += 16) {
        float p = __expf(myrow[c] - rmax);
        myrow[c] = p;
        s += p;
    }
    __syncthreads();
    red[srow * 16 + scol0] = s;
    __syncthreads();
    if (tid < 16) {
        float ss = 0.f;
        #pragma unroll
        for (int i = 0; i < 16; ++i) ss += red[tid * 16 + i];
        rowstat[tid] = 1.0f / ss;
    }
    __syncthreads();
    float rinv = rowstat[srow];
    for (int c = scol0; c < SEQ; c += 16) myrow[c] *= rinv;
    __syncthreads();

    // ======== Phase 3: stream attn tile to HBM (nontemporal, 1 GB total) ========
    {
        float* attn_tile = attn + (size_t)(h * SEQ + rt * ROWS) * SEQ;
        for (int idx = tid; idx < ROWS * SEQ; idx += 256) {
            int row = idx >> 12;      // /4096
            int col = idx & 4095;
            __builtin_nontemporal_store(sc[row * SC_STRIDE + col], attn_tile + idx);
        }
    }

    // ================= Phase 4: O = P @ V (k split across waves) =================
    v8f oacc[4] = {};
    for (int vt = wave * CT_PER_WAVE; vt < wave * CT_PER_WAVE + CT_PER_WAVE; ++vt) {
        {   // stage V tile [16 k-rows][64 dims]
            const float4* src = (const float4*)(vh + (size_t)vt * 16 * DDIM);
            #pragma unroll
            for (int i = 0; i < 8; ++i) {
                int j = lane + 32 * i;
                int r = j >> 4, c4 = j & 15;
                float4 d = src[j];
                float* dst = mystg + r * ST_STRIDE + c4 * 4;
                dst[0] = d.x; dst[1] = d.y; dst[2] = d.z; dst[3] = d.w;
            }
        }
        int kbase = vt * 16;
        #pragma unroll
        for (int kc = 0; kc < 4; ++kc) {
            int kl = 4 * kc + 2 * half;
            v2f a;   // A = P fragment: M=ln16, K pair {kl, kl+1}
            const float* ap = sc + ln16 * SC_STRIDE + kbase + kl;
            a.x = ap[0]; a.y = ap[1];
            #pragma unroll
            for (int nt = 0; nt < 4; ++nt) {
                v2f b;   // B = V: B[kk][n] = V[kbase+kl+kk][nt*16+n]
                b.x = mystg[kl * ST_STRIDE + nt * 16 + ln16];
                b.y = mystg[(kl + 1) * ST_STRIDE + nt * 16 + ln16];
                oacc[nt] = __builtin_amdgcn_wmma_f32_16x16x4_f32(
                    false, a, false, b, (short)0, oacc[nt], false, false);
            }
        }
    }
    __syncthreads();   // everyone done reading sc -> safe to reuse as partial buffer

    // per-wave partial O tiles into LDS, then cross-wave reduce + store
    {
        float* part = sc + wave * (ROWS * DDIM);
        #pragma unroll
        for (int nt = 0; nt < 4; ++nt)
            #pragma unroll
            for (int j = 0; j < 8; ++j) {
                int row = j + 8 * half;
                part[row * DDIM + nt * 16 + ln16] = oacc[nt][j];
            }
    }
    __syncthreads();
    {
        float* outp = out + headOff + (size_t)rt * ROWS * DDIM;
        for (int e = tid; e < ROWS * DDIM; e += 256) {
            float ssum = 0.f;
            #pragma unroll
            for (int w = 0; w < NWAVE; ++w) ssum += sc[w * (ROWS * DDIM) + e];
            outp[e] = ssum;
        }
    }
}

extern "C" void kernel_launch(void* const* d_in, const int* in_sizes, int n_in,
                              void* d_out, int out_size, void* d_ws, size_t ws_size,
                              hipStream_t stream) {
    const float* q    = (const float*)d_in[0];
    const float* k    = (const float*)d_in[1];
    const float* v    = (const float*)d_in[2];
    const int*   mask = (const int*)d_in[3];
    float* out  = (float*)d_out;
    float* attn = out + (size_t)HDIM * SEQ * DDIM;   // tuple: out first, then attn

    size_t shmem = (size_t)(ROWS * SC_STRIDE + NWAVE * 16 * ST_STRIDE + 256 + 16)
                   * sizeof(float);                  // ~298 KB of the WGP's 320 KB LDS
    (void)hipFuncSetAttribute((const void*)attn_fused_kernel,
                              hipFuncAttributeMaxDynamicSharedMemorySize, (int)shmem);

    dim3 grid(HDIM, SEQ / ROWS);   // (16 heads, 256 row tiles)
    hipLaunchKernelGGL(attn_fused_kernel, grid, dim3(256), shmem, stream,
                       q, k, v, mask, out, attn);
}